// BiGraphConv_88725434401306
// MI455X (gfx1250) — compile-verified
//
#include <hip/hip_runtime.h>

typedef __attribute__((ext_vector_type(2))) float v2f;
typedef __attribute__((ext_vector_type(8))) float v8f;

#define BF 128   // b_features (K of GEMM)
#define AF 128   // a_features (N of GEMM)

// ---------------------------------------------------------------------------
// Kernel 1: a_support[NB, AF] = b_input[NB, BF] @ a_weight[BF, AF]
// One block = 16 rows of b_input; 8 waves, each wave computes a 16x16 tile
// using V_WMMA_F32_16X16X4_F32, accumulating over K=128 in steps of 4.
// ---------------------------------------------------------------------------
__global__ __launch_bounds__(256) void bigc_gemm_wmma_f32(
    const float* __restrict__ bin,   // [NB, BF]
    const float* __restrict__ w,     // [BF, AF]
    float* __restrict__ sup,         // [NB, AF] (workspace)
    int nb_rows) {
  const int lane = threadIdx.x & 31;
  const int wv   = threadIdx.x >> 5;        // wave 0..7 -> N tile
  const int row0 = blockIdx.x << 4;         // 16 rows per block
  if (row0 >= nb_rows) return;              // block-uniform, EXEC stays full
  const int n0 = wv << 4;                   // feature tile base
  const int m  = lane & 15;                 // row/col within tile
  const int kh = (lane >> 4) << 1;          // 0 or 2: K sub-offset per A/B layout

  const float* arow = bin + (size_t)(row0 + m) * BF;   // A: row m of tile
  const float* bcol = w + n0 + m;                      // B: col m of tile

  v8f acc = {};
  #pragma unroll 4
  for (int k = 0; k < BF; k += 4) {
    // A frag (16x4 f32): lane L vgpr j -> A[L%16][j + 2*(L/16)]
    v2f a;
    a[0] = arow[k + kh];
    a[1] = arow[k + kh + 1];
    // B frag (4x16 f32): lane L vgpr j -> B[j + 2*(L/16)][L%16]
    v2f b;
    b[0] = bcol[(size_t)(k + kh) * AF];
    b[1] = bcol[(size_t)(k + kh + 1) * AF];
    // 8 args: (neg_a, A, neg_b, B, c_mod, C, reuse_a, reuse_b)
    acc = __builtin_amdgcn_wmma_f32_16x16x4_f32(
        false, a, false, b, (short)0, acc, false, false);
  }

  // C/D layout: VGPR r, lane L -> row (r + 8*(L/16)), col L%16
  float* orow = sup + (size_t)row0 * AF + n0;
  const int mhi = (lane >> 4) << 3;
  #pragma unroll
  for (int r = 0; r < 8; ++r)
    orow[(size_t)(r + mhi) * AF + m] = acc[r];
}

// ---------------------------------------------------------------------------
// Kernel 2: a_output[r, f] = a_bias[f]  (initializes d_out; scatter adds on top)
// ---------------------------------------------------------------------------
__global__ void bigc_bias_init(float* __restrict__ out,
                               const float* __restrict__ bias,
                               int total) {
  int i = blockIdx.x * blockDim.x + threadIdx.x;
  if (i < total) out[i] = bias[i & (AF - 1)];
}

// ---------------------------------------------------------------------------
// Hardware fp32 atomic add, no return value (tracked by STOREcnt; the implicit
// S_WAIT_IDLE at S_ENDPGM drains it). Guarantees global_atomic_add_f32 — the
// L2 atomic unit path — rather than a CAS replay loop.
// ---------------------------------------------------------------------------
__device__ __forceinline__ void atomic_add_f32_hw(float* addr, float val) {
  asm volatile("global_atomic_add_f32 %0, %1, off"
               :: "v"(addr), "v"(val)
               : "memory");
}

// ---------------------------------------------------------------------------
// Kernel 3: for each edge e: a_output[rows[e], :] += vals[e] * a_support[cols[e], :]
// One wave per edge; lane handles 4 consecutive features (float4 gather,
// 4x global_atomic_add_f32). Output (51 MB) is L2-resident on MI455X (192 MB).
// ---------------------------------------------------------------------------
__global__ __launch_bounds__(256) void bigc_spmm_scatter(
    const float* __restrict__ sup,   // [NB, AF]
    const int* __restrict__ rows,    // [NE]
    const int* __restrict__ cols,    // [NE]
    const float* __restrict__ vals,  // [NE]
    float* __restrict__ out,         // [NA, AF]
    int ne) {
  const int lane = threadIdx.x & 31;
  int wid = (int)((blockIdx.x * blockDim.x + threadIdx.x) >> 5);
  int nw  = (int)((gridDim.x * blockDim.x) >> 5);
  for (int e = wid; e < ne; e += nw) {
    const int   r = rows[e];
    const int   c = cols[e];
    const float v = vals[e];
    const float4* srow = (const float4*)(sup + (size_t)c * AF);
    float4 d = srow[lane];                    // global_load_b128 gather
    d.x *= v; d.y *= v; d.z *= v; d.w *= v;
    float* orow = out + (size_t)r * AF + (lane << 2);
    atomic_add_f32_hw(orow + 0, d.x);
    atomic_add_f32_hw(orow + 1, d.y);
    atomic_add_f32_hw(orow + 2, d.z);
    atomic_add_f32_hw(orow + 3, d.w);
  }
}

// ---------------------------------------------------------------------------
// inputs: [0] b_input f32 [NB,128], [1] edge_rows i32 [NE], [2] edge_cols i32 [NE],
//         [3] edge_vals f32 [NE],   [4] a_weight f32 [128,128], [5] a_bias f32 [128]
// output: a_output f32 [NA,128];  d_ws: a_support f32 [NB,128] (51.2 MB)
// ---------------------------------------------------------------------------
extern "C" void kernel_launch(void* const* d_in, const int* in_sizes, int n_in,
                              void* d_out, int out_size, void* d_ws, size_t ws_size,
                              hipStream_t stream) {
  const float* b_input   = (const float*)d_in[0];
  const int*   edge_rows = (const int*)  d_in[1];
  const int*   edge_cols = (const int*)  d_in[2];
  const float* edge_vals = (const float*)d_in[3];
  const float* a_weight  = (const float*)d_in[4];
  const float* a_bias    = (const float*)d_in[5];
  float* out = (float*)d_out;
  float* sup = (float*)d_ws;

  const int nb = in_sizes[0] / BF;   // 100000
  const int ne = in_sizes[1];        // 1600000
  const int na = out_size / AF;      // 100000

  // 1) GEMM into workspace
  bigc_gemm_wmma_f32<<<(nb + 15) / 16, 256, 0, stream>>>(b_input, a_weight, sup, nb);

  // 2) out = broadcast(bias)
  const int total = na * AF;
  bigc_bias_init<<<(total + 255) / 256, 256, 0, stream>>>(out, a_bias, total);

  // 3) edge scatter with hardware float atomics (stream order guarantees 1,2 done)
  bigc_spmm_scatter<<<4096, 256, 0, stream>>>(sup, edge_rows, edge_cols, edge_vals,
                                              out, ne);
}